// Quantizer_84430467105116
// MI455X (gfx1250) — compile-verified
//
#include <hip/hip_runtime.h>

typedef __attribute__((ext_vector_type(2))) float v2f;
typedef __attribute__((ext_vector_type(8))) float v8f;

#define NUM_EMB   512
#define EMB_DIM   64
#define N_ROWS    131072      // 32*64*64 flattened rows
#define LDSS      528         // padded code stride for transposed codebook in LDS
#define Q_ELEMS   8388608     // 32*64*64*64
#define LOSS_OFF  8388608     // d_out offsets: [quantized | loss | encodings | perplexity]
#define ENC_OFF   8388609
#define PERP_OFF  75497473

// ---------------------------------------------------------------------------
// Zero the 256MB one-hot encodings region + workspace accumulators.
// Region start (float index 8388608) is 16B-aligned -> float4 stores.
// ---------------------------------------------------------------------------
__global__ void vq_zero(float* __restrict__ out, float* __restrict__ ws) {
    const long n4 = 16777216;             // 67,108,864 floats / 4
    float4* p = (float4*)out;
    const long base4 = (long)LOSS_OFF / 4;
    long i = (long)blockIdx.x * blockDim.x + threadIdx.x;
    const long stride = (long)gridDim.x * blockDim.x;
    float4 z; z.x = 0.f; z.y = 0.f; z.z = 0.f; z.w = 0.f;
    for (long k = i; k < n4; k += stride) p[base4 + k] = z;
    if (i < 513) ws[i] = 0.0f;            // ws[0]=loss accum, ws[1..512]=counts
    if (i == 0) { out[75497472] = 0.0f; out[PERP_OFF] = 0.0f; } // tail floats
}

// ---------------------------------------------------------------------------
// Main kernel: per-wave 16-row tile, full 512-code scan via f32 WMMA.
// ---------------------------------------------------------------------------
__global__ __launch_bounds__(256) void vq_main(
    const float* __restrict__ x, const float* __restrict__ w,
    float* __restrict__ qout, float* __restrict__ enc,
    float* __restrict__ lossAcc, float* __restrict__ counts)
{
    __shared__ float wT[EMB_DIM * LDSS];   // transposed codebook, padded stride
    __shared__ float wsq[NUM_EMB];         // per-code squared norms
    __shared__ int   sIdx[8][16];          // per-wave winning indices

    const int tid = threadIdx.x;

    // Stage w^T into LDS (coalesced global reads of w row-major).
    for (int i = tid; i < NUM_EMB * EMB_DIM; i += 256) {
        int code = i >> 6, feat = i & 63;
        wT[feat * LDSS + code] = w[i];
    }
    __syncthreads();
    // Per-code squared norms.
    for (int c = tid; c < NUM_EMB; c += 256) {
        float s = 0.f;
        #pragma unroll 8
        for (int f = 0; f < EMB_DIM; ++f) { float v = wT[f * LDSS + c]; s += v * v; }
        wsq[c] = s;
    }
    __syncthreads();

    const int wave = tid >> 5;
    const int lane = tid & 31;
    const int half = lane >> 4;            // 0: lanes 0-15, 1: lanes 16-31
    const int lpos = lane & 15;
    const int row0 = (blockIdx.x * 8 + wave) * 16;

    // Row n = b*4096 + h*64 + w ; element (n,d) lives at d*4096 + n + b*258048.
    const int rowA  = row0 + lpos;
    const int baseA = rowA + (rowA >> 12) * 258048;

    // A-tile: 16 rows x 64 feats, WMMA 16x4 f32 layout.
    // VGPR pair kb: lanes 0-15 -> feats {4kb,4kb+1}, lanes 16-31 -> {4kb+2,4kb+3}.
    v2f a[16];
    #pragma unroll
    for (int kb = 0; kb < 16; ++kb) {
        const int f0 = kb * 4 + half * 2;
        a[kb].x = x[baseA + f0 * 4096];
        a[kb].y = x[baseA + (f0 + 1) * 4096];
    }

    // B operand lane base: lanes 0-15 read feat f, lanes 16-31 read feat f+2.
    const int bOff = half * 2 * LDSS + lpos;

    float bestv[8];
    int   besti[8];
    #pragma unroll
    for (int j = 0; j < 8; ++j) { bestv[j] = 3.4e38f; besti[j] = 0; }

    for (int cb = 0; cb < 32; ++cb) {      // 32 blocks of 16 codes
        const int c0 = cb * 16;
        v8f acc = {};
        #pragma unroll
        for (int kb = 0; kb < 16; ++kb) {  // K = 64 = 16 x 4
            const float* bp = &wT[kb * 4 * LDSS + bOff + c0];
            v2f b;
            b.x = bp[0];
            b.y = bp[LDSS];
            acc = __builtin_amdgcn_wmma_f32_16x16x4_f32(
                      false, a[kb], false, b, (short)0, acc, false, false);
        }
        // distance (up to per-row constant) = ||w||^2 - 2 x.w
        const float wq = wsq[c0 + lpos];
        #pragma unroll
        for (int j = 0; j < 8; ++j) {
            float dist = wq - 2.0f * acc[j];
            if (dist < bestv[j]) { bestv[j] = dist; besti[j] = c0 + lpos; }
        }
    }

    // Argmin reduce across the 16 lanes sharing each row (prefer lower index on tie).
    #pragma unroll
    for (int j = 0; j < 8; ++j) {
        #pragma unroll
        for (int m = 1; m <= 8; m <<= 1) {
            float ov = __shfl_xor(bestv[j], m, 32);
            int   oi = __shfl_xor(besti[j], m, 32);
            if (ov < bestv[j] || (ov == bestv[j] && oi < besti[j])) {
                bestv[j] = ov; besti[j] = oi;
            }
        }
    }

    // C layout: vgpr j holds row j (lanes 0-15) and row j+8 (lanes 16-31).
    if (lpos == 0) {
        #pragma unroll
        for (int j = 0; j < 8; ++j) sIdx[wave][j + 8 * half] = besti[j];
    }
    __syncthreads();

    const int myIdx = sIdx[wave][lpos];    // winning code for row rowA

    // One-hot scatter + histogram (one lane per row).
    if (half == 0) {
        enc[(long)rowA * NUM_EMB + myIdx] = 1.0f;
        atomicAdd(&counts[myIdx], 1.0f);
    }

    // Quantized output (straight-through forward value) + MSE accumulation.
    float lsum = 0.0f;
    #pragma unroll
    for (int kb = 0; kb < 16; ++kb) {
        const int f0 = kb * 4 + half * 2;
        float q0 = wT[f0 * LDSS + myIdx];
        float q1 = wT[(f0 + 1) * LDSS + myIdx];
        float x0 = a[kb].x, x1 = a[kb].y;
        float d0 = q0 - x0, d1 = q1 - x1;
        lsum += d0 * d0 + d1 * d1;
        qout[baseA + f0 * 4096]       = x0 + d0;   // == x + (q - x), as in reference
        qout[baseA + (f0 + 1) * 4096] = x1 + d1;
    }
    #pragma unroll
    for (int m = 1; m < 32; m <<= 1) lsum += __shfl_xor(lsum, m, 32);
    if (lane == 0) atomicAdd(lossAcc, lsum);
}

// ---------------------------------------------------------------------------
// Finalize: perplexity from histogram, scalar loss.
// ---------------------------------------------------------------------------
__global__ void vq_finalize(const float* __restrict__ ws, float* __restrict__ out) {
    __shared__ float red[512];
    const int t = threadIdx.x;
    float c = ws[1 + t];
    float p = c * (1.0f / (float)N_ROWS);
    red[t] = p * logf(p + 1e-10f);
    __syncthreads();
    for (int s = 256; s > 0; s >>= 1) {
        if (t < s) red[t] += red[t + s];
        __syncthreads();
    }
    if (t == 0) {
        out[PERP_OFF] = expf(-red[0]);
        out[LOSS_OFF] = 1.25f * ws[0] * (1.0f / (float)Q_ELEMS);  // q + 0.25*e, q==e
    }
}

// ---------------------------------------------------------------------------
extern "C" void kernel_launch(void* const* d_in, const int* in_sizes, int n_in,
                              void* d_out, int out_size, void* d_ws, size_t ws_size,
                              hipStream_t stream) {
    const float* x = (const float*)d_in[0];   // [32,64,64,64] f32
    const float* w = (const float*)d_in[1];   // [512,64] f32
    float* out = (float*)d_out;
    float* ws  = (float*)d_ws;

    vq_zero<<<2048, 256, 0, stream>>>(out, ws);
    vq_main<<<1024, 256, 0, stream>>>(x, w,
                                      out,                 // quantized_out
                                      out + ENC_OFF,       // encodings
                                      ws,                  // loss accumulator
                                      ws + 1);             // counts[512]
    vq_finalize<<<1, 512, 0, stream>>>(ws, out);
}